// CapsuleLayer_89575837925793
// MI455X (gfx1250) — compile-verified
//
#include <hip/hip_runtime.h>
#include <math.h>
#include <stdint.h>

// CDNA5 / gfx1250 capsule dynamic-routing kernel.
// - All three einsums on V_WMMA_F32_16X16X4_F32 (wave32 WMMA, full f32).
// - routing_logits_init streamed Global->LDS with GLOBAL_LOAD_ASYNC_TO_LDS_B128
//   (ASYNCcnt) overlapped with the main GEMM.
// - Non-temporal loads for the 256 MB one-touch input stream, non-temporal
//   stores for the outputs; bilinear matrix stays cache-resident.

typedef __attribute__((ext_vector_type(2))) float v2f;
typedef __attribute__((ext_vector_type(8))) float v8f;

#define BATCHN   1024
#define SEQN     256
#define LOWD     256
#define HIGHD    128
#define MAXK     16
#define NITERS   3
#define LOWP     132   // padded LDS row stride for low  (avoids bank conflicts)
#define HIGHP    132   // padded LDS row stride for high (avoids bank conflicts)

__device__ __forceinline__ v8f wmma4(v2f a, v2f b, v8f c) {
  // 8-arg form: (neg_a, A, neg_b, B, c_mod, C, reuse_a, reuse_b)
  return __builtin_amdgcn_wmma_f32_16x16x4_f32(false, a, false, b, (short)0, c,
                                               false, false);
}

__launch_bounds__(256, 1)
__global__ void capsule_routing_kernel(const float* __restrict__ inputs,   // [B,S,L]
                                       const int*   __restrict__ seq_len,  // [B]
                                       const float* __restrict__ Wmat,     // [L,H]
                                       const float* __restrict__ rl_init,  // [B,S,K]
                                       float* __restrict__ out_ui,         // [B,K,H]
                                       float* __restrict__ out_mask) {     // [B,K]
  extern __shared__ float smem[];
  float* lds_low   = smem;                       // SEQN*LOWP
  float* lds_p     = lds_low  + SEQN * LOWP;     // SEQN*MAXK
  float* lds_rl    = lds_p    + SEQN * MAXK;     // SEQN*MAXK
  float* lds_high  = lds_rl   + SEQN * MAXK;     // MAXK*HIGHP
  float* lds_inv   = lds_high + MAXK * HIGHP;    // SEQN
  float* lds_scale = lds_inv  + SEQN;            // MAXK

  const int b    = blockIdx.x;
  const int t    = threadIdx.x;
  const int w    = t >> 5;       // wave id (8 waves)
  const int lane = t & 31;
  const int lm   = lane & 15;    // M/N index within 16x16 tile
  const int hl   = lane >> 4;    // half-wave select

  // ---- Kick off async Global->LDS copy of this batch's routing-logit rows.
  // Each thread copies its 64-byte row (s = t) with 4x B128 async transfers.
  // IOFFSET applies to BOTH the LDS and global side (CDNA5 ISA).
  {
    const uint32_t ldsa = (uint32_t)(uintptr_t)lds_rl + (uint32_t)t * 64u;
    const uint32_t va   = (uint32_t)t * 64u;  // per-thread byte offset
    const uint64_t sa   = (uint64_t)(uintptr_t)(rl_init + (size_t)b * SEQN * MAXK);
    asm volatile(
        "global_load_async_to_lds_b128 %0, %1, %2\n\t"
        "global_load_async_to_lds_b128 %0, %1, %2 offset:16\n\t"
        "global_load_async_to_lds_b128 %0, %1, %2 offset:32\n\t"
        "global_load_async_to_lds_b128 %0, %1, %2 offset:48"
        :: "v"(ldsa), "v"(va), "s"(sa)
        : "memory");
  }

  const int   len   = seq_len[b];
  const float nhigh = fmaxf(1.0f, fminf((float)MAXK, log2f((float)len)));
  unsigned act = 0;
  #pragma unroll
  for (int k = 0; k < MAXK; ++k)
    if ((float)k < nhigh) act |= (1u << k);

  const v8f vzero = {0.f, 0.f, 0.f, 0.f, 0.f, 0.f, 0.f, 0.f};

  // ---------------- Phase 1: low = masked(inputs[b]) @ W  (256x256 x 256x128)
  const float* A = inputs + (size_t)b * SEQN * LOWD;
  for (int rtp = 0; rtp < 2; ++rtp) {
    const int   rt    = w * 2 + rtp;             // row tile 0..15
    const int   row   = rt * 16 + lm;            // A-fragment row (lanes 0-15 = M)
    const float rmask = (row < len) ? 1.0f : 0.0f;
    const float* Arow = A + (size_t)row * LOWD;

    v8f acc[8];
    #pragma unroll
    for (int ct = 0; ct < 8; ++ct) acc[ct] = vzero;

    for (int kk = 0; kk < LOWD; kk += 4) {
      if (kk + 64 < LOWD) __builtin_prefetch(Arow + kk + 64, 0, 1); // global_prefetch_b8
      // A 16x4 f32 layout: K = 2*half + vgpr. One-touch stream -> non-temporal.
      v2f a = __builtin_nontemporal_load((const v2f*)(Arow + kk + 2 * hl));
      a *= rmask;
      #pragma unroll
      for (int ct = 0; ct < 8; ++ct) {
        const int col = ct * 16 + lm;
        v2f bb;                                   // reused matrix: keep cacheable
        bb.x = Wmat[(size_t)(kk + 2 * hl)     * HIGHD + col];
        bb.y = Wmat[(size_t)(kk + 2 * hl + 1) * HIGHD + col];
        acc[ct] = wmma4(a, bb, acc[ct]);
      }
    }
    // C layout: VGPR j -> M = j + 8*half, N = lm
    #pragma unroll
    for (int ct = 0; ct < 8; ++ct)
      #pragma unroll
      for (int j = 0; j < 8; ++j)
        lds_low[(size_t)(rt * 16 + j + 8 * hl) * LOWP + ct * 16 + lm] = acc[ct][j];
  }
  __syncthreads();

  // ---------------- Phase 2: row inverse norms (rl already streaming to LDS)
  {
    const int s = t;  // one row per thread
    float sum = 0.f;
    #pragma unroll 4
    for (int h = 0; h < HIGHD; ++h) {
      const float v = lds_low[s * LOWP + h];
      sum += v * v;
    }
    lds_inv[s] = 1.0f / fmaxf(sqrtf(sum), 1e-12f);
  }
  // Our wave's async LDS writes must land before the barrier publishes them.
  asm volatile("s_wait_asynccnt 0" ::: "memory");
  __syncthreads();

  // ---------------- Routing iterations
  for (int it = 0; it < NITERS; ++it) {
    // (a) softmax over active capsules, then seq-mask. One row per thread.
    {
      const int s = t;
      float v[MAXK];
      float mx = -3.0e38f;
      #pragma unroll
      for (int k = 0; k < MAXK; ++k) {
        const float x = ((act >> k) & 1u) ? lds_rl[s * MAXK + k] : -3.0e38f;
        v[k] = x;
        mx = fmaxf(mx, x);
      }
      float sum = 0.f;
      #pragma unroll
      for (int k = 0; k < MAXK; ++k) {
        const float e = ((act >> k) & 1u) ? __expf(v[k] - mx) : 0.0f;
        v[k] = e;
        sum += e;
      }
      const float sc = (s < len) ? (1.0f / sum) : 0.0f;
      #pragma unroll
      for (int k = 0; k < MAXK; ++k) lds_p[s * MAXK + k] = v[k] * sc;
    }
    __syncthreads();

    // (b) high[k,h] = sum_s p[s,k] * low[s,h]  -> 16x128, wave w owns H-cols [16w,16w+16)
    {
      v8f acc = vzero;
      for (int ss = 0; ss < SEQN; ss += 4) {
        v2f a;                       // A[m=k][kk=s] = p[s][k]
        a.x = lds_p[(ss + 2 * hl)     * MAXK + lm];
        a.y = lds_p[(ss + 2 * hl + 1) * MAXK + lm];
        v2f bb;                      // B[kk=s][n=h] = low[s][16w+n]
        bb.x = lds_low[(size_t)(ss + 2 * hl)     * LOWP + w * 16 + lm];
        bb.y = lds_low[(size_t)(ss + 2 * hl + 1) * LOWP + w * 16 + lm];
        acc = wmma4(a, bb, acc);
      }
      #pragma unroll
      for (int j = 0; j < 8; ++j)
        lds_high[(j + 8 * hl) * HIGHP + w * 16 + lm] = acc[j];
    }
    __syncthreads();

    // (c) rl[s,k] = p[s,k] + inv[s] * sum_h low[s,h] * high[k,h]   (iters 0,1 only)
    if (it + 1 < NITERS) {
      for (int rtp = 0; rtp < 2; ++rtp) {
        const int rt = w * 2 + rtp;
        v8f acc = vzero;
        for (int hh = 0; hh < HIGHD; hh += 4) {
          v2f a;                     // A[m=s][kk=h] = low[rt*16+m][h]
          a.x = lds_low[(size_t)(rt * 16 + lm) * LOWP + hh + 2 * hl];
          a.y = lds_low[(size_t)(rt * 16 + lm) * LOWP + hh + 2 * hl + 1];
          v2f bb;                    // B[kk=h][n=k] = high[k][h]  (transposed read, padded)
          bb.x = lds_high[lm * HIGHP + hh + 2 * hl];
          bb.y = lds_high[lm * HIGHP + hh + 2 * hl + 1];
          acc = wmma4(a, bb, acc);
        }
        #pragma unroll
        for (int j = 0; j < 8; ++j) {
          const int srow = rt * 16 + j + 8 * hl;
          lds_rl[srow * MAXK + lm] =
              lds_p[srow * MAXK + lm] + acc[j] * lds_inv[srow];
        }
      }
      __syncthreads();
    }
  }
  __syncthreads();

  // ---------------- Squash + capsule mask + store
  if (t < MAXK) {
    float sum = 0.f;
    #pragma unroll 4
    for (int h = 0; h < HIGHD; ++h) {
      const float v = lds_high[t * HIGHP + h];
      sum += v * v;
    }
    const float ne = fmaxf(sqrtf(sum), 1e-7f);
    // scale = (ne^2/(1+ne^2))^1 / ne = ne/(1+ne^2)
    lds_scale[t] = ((act >> t) & 1u) ? (ne / (1.0f + ne * ne)) : 0.0f;
  }
  __syncthreads();

  float* outb = out_ui + (size_t)b * MAXK * HIGHD;
  for (int e = t; e < MAXK * HIGHD; e += 256) {
    const int k = e >> 7;
    const int h = e & 127;
    __builtin_nontemporal_store(lds_high[k * HIGHP + h] * lds_scale[k], outb + e);
  }
  if (t < MAXK)
    __builtin_nontemporal_store(((act >> t) & 1u) ? 1.0f : 0.0f,
                                out_mask + (size_t)b * MAXK + t);
}

extern "C" void kernel_launch(void* const* d_in, const int* in_sizes, int n_in,
                              void* d_out, int out_size, void* d_ws, size_t ws_size,
                              hipStream_t stream) {
  const float* inputs  = (const float*)d_in[0];   // [1024,256,256] f32
  const int*   seq_len = (const int*)d_in[1];     // [1024] i32
  const float* Wmat    = (const float*)d_in[2];   // [256,128] f32
  const float* rl_init = (const float*)d_in[3];   // [1024,256,16] f32
  float* out_ui   = (float*)d_out;                              // [1024,16,128]
  float* out_mask = out_ui + (size_t)BATCHN * MAXK * HIGHD;     // [1024,16]

  const size_t shmem =
      (size_t)(SEQN * LOWP + 2 * SEQN * MAXK + MAXK * HIGHP + SEQN + MAXK) *
      sizeof(float);  // ~177 KB of the 320 KB WGP LDS

  (void)hipFuncSetAttribute(reinterpret_cast<const void*>(capsule_routing_kernel),
                            hipFuncAttributeMaxDynamicSharedMemorySize,
                            (int)shmem);

  capsule_routing_kernel<<<BATCHN, 256, shmem, stream>>>(
      inputs, seq_len, Wmat, rl_init, out_ui, out_mask);
}